// _SmoothMatrixDecompositionBase_3693671875225
// MI455X (gfx1250) — compile-verified
//
#include <hip/hip_runtime.h>
#include <hip/hip_bf16.h>

typedef float v2f __attribute__((ext_vector_type(2)));
typedef float v8f __attribute__((ext_vector_type(8)));

#define BSZ 16      // B*S
#define DD  160     // D = T/S
#define NN  20736   // C*H*W
#define PP  447     // P2 (RBF columns)
#define PPQ 448     // padded P2
#define EPSV 1e-6f

// ---------------- init kernels ----------------

__global__ void k_rbf(float* __restrict__ rbf) {
    int i = blockIdx.x * 256 + threadIdx.x;
    if (i >= DD * PPQ) return;
    int d = i / PPQ, p = i % PPQ;
    float val;
    if (p >= PP)       val = 0.0f;           // pad column
    else if (p == 446) val = 1.0f;           // ones column
    else {
        float sig, c;
        if      (p < 160) { sig = 1.0f;  c = (float)p; }
        else if (p < 240) { sig = 2.0f;  c = (float)(2 * (p - 160)); }
        else if (p < 320) { sig = 4.0f;  c = (float)(2 * (p - 240)); }
        else if (p < 374) { sig = 6.0f;  c = (float)(3 * (p - 320)); }
        else if (p < 414) { sig = 8.0f;  c = (float)(4 * (p - 374)); }
        else              { sig = 10.0f; c = (float)(5 * (p - 414)); }
        float t = (float)d - c;
        val = expf(-0.25f * t * t / (sig * sig));   // exp(-0.5*dd/(2*sig^2))
    }
    rbf[i] = val;
}

__global__ void k_initb(const float* __restrict__ bases0, float* __restrict__ basesP) {
    int i = blockIdx.x * 256 + threadIdx.x;
    if (i >= PPQ * BSZ) return;
    int p = i / BSZ, b = i % BSZ;
    basesP[i] = (p < PP) ? bases0[b * PP + p] : 0.0f;  // [p][b] layout, zero pad row
}

__global__ void k_initc(float* __restrict__ coef) {
    size_t i = (size_t)blockIdx.x * 256 + threadIdx.x;
    if (i < (size_t)BSZ * NN) coef[i] = 1.0f;          // softmax over R=1 -> 1
}

// ---------------- S1: bp = RBF(160x448) x basesP(448x16) via fp32 WMMA ----------------

__global__ __launch_bounds__(320) void k_s1(const float* __restrict__ rbf,
                                            const float* __restrict__ basesP,
                                            float* __restrict__ bp) {
    int tid = threadIdx.x;
    int wave = tid >> 5;            // 10 waves, one 16-row M tile each
    int lane = tid & 31;
    int m0 = wave * 16;
    int mn = lane & 15;             // A row within tile == B/D column
    int kh = (lane >> 4) << 1;      // 0 or 2 (K pair)
    v8f acc = {};
    for (int k0 = 0; k0 < PPQ; k0 += 4) {
        v2f a, b;
        a.x = rbf[(m0 + mn) * PPQ + (k0 + kh)];
        a.y = rbf[(m0 + mn) * PPQ + (k0 + kh + 1)];
        b.x = basesP[(k0 + kh) * BSZ + mn];
        b.y = basesP[(k0 + kh + 1) * BSZ + mn];
        acc = __builtin_amdgcn_wmma_f32_16x16x4_f32(false, a, false, b,
                                                    (short)0, acc, false, false);
    }
    int rbase = m0 + ((lane >> 4) << 3);   // lanes>=16 hold rows M+8
#pragma unroll
    for (int v = 0; v < 8; ++v)
        bp[(rbase + v) * BSZ + mn] = acc[v];
}

// ---------------- NC: coef <- coef*num_c/(coef*||bp||^2 + eps) ----------------

__global__ __launch_bounds__(256) void k_nc(const float* __restrict__ x,
                                            const float* __restrict__ bp,
                                            float* __restrict__ coef) {
    __shared__ float lbp[DD];
    int b = blockIdx.y, tid = threadIdx.x;
    if (tid < DD) lbp[tid] = bp[tid * BSZ + b];
    __syncthreads();
    int n = blockIdx.x * 256 + tid;                 // 81*256 == NN exactly
    const float* xb = x + (size_t)b * DD * NN;
    float acc = 0.0f, sbp = 0.0f;
#pragma unroll 4
    for (int d = 0; d < DD; ++d) {
        float v = lbp[d];
        acc += xb[(size_t)d * NN + n] * v;
        sbp += v * v;
    }
    size_t ci = (size_t)b * NN + n;
    float c = coef[ci];
    coef[ci] = c * acc / (c * sbp + EPSV);
}

// ---------------- SC: s_c[b] = ||coef_b||^2 ----------------

__global__ __launch_bounds__(256) void k_sc(const float* __restrict__ coef,
                                            float* __restrict__ s_c) {
    __shared__ float red[256];
    int b = blockIdx.x, tid = threadIdx.x;
    const float* cr = coef + (size_t)b * NN;
    float acc = 0.0f;
    for (int n = tid; n < NN; n += 256) { float c = cr[n]; acc += c * c; }
    red[tid] = acc; __syncthreads();
    for (int s = 128; s > 0; s >>= 1) { if (tid < s) red[tid] += red[tid + s]; __syncthreads(); }
    if (tid == 0) s_c[b] = red[0];
}

// ---------------- XC: xc[d][b] = sum_n x[b][d][n]*coef[b][n] ----------------

__global__ __launch_bounds__(256) void k_xc(const float* __restrict__ x,
                                            const float* __restrict__ coef,
                                            float* __restrict__ xc) {
    __shared__ float red[256];
    int d = blockIdx.x, b = blockIdx.y, tid = threadIdx.x;
    const float* xr = x + ((size_t)b * DD + d) * NN;
    const float* cr = coef + (size_t)b * NN;
    float acc = 0.0f;
    for (int n = tid; n < NN; n += 256) acc += xr[n] * cr[n];
    red[tid] = acc; __syncthreads();
    for (int s = 128; s > 0; s >>= 1) { if (tid < s) red[tid] += red[tid + s]; __syncthreads(); }
    if (tid == 0) xc[d * BSZ + b] = red[0];
}

// ---------------- S2: [num_b|rtrb] = RBF^T(448x160) x [xc|bp](160x32); bases update ----

__global__ __launch_bounds__(896) void k_s2(const float* __restrict__ rbf,
                                            const float* __restrict__ xc,
                                            const float* __restrict__ bp,
                                            const float* __restrict__ s_c,
                                            float* __restrict__ basesP) {
    int tid = threadIdx.x;
    int wave = tid >> 5;            // 28 waves, one 16-row p-tile each
    int lane = tid & 31;
    int m0 = wave * 16;
    int mn = lane & 15;
    int kh = (lane >> 4) << 1;
    v8f acc0 = {};                  // num_b tile
    v8f acc1 = {};                  // rtrb tile
    for (int k0 = 0; k0 < DD; k0 += 4) {
        int ka = k0 + kh;
        v2f a, b0, b1;
        a.x  = rbf[ka * PPQ + (m0 + mn)];       // RBF^T[p][d] = rbf[d*PPQ+p]
        a.y  = rbf[(ka + 1) * PPQ + (m0 + mn)];
        b0.x = xc[ka * BSZ + mn];  b0.y = xc[(ka + 1) * BSZ + mn];
        b1.x = bp[ka * BSZ + mn];  b1.y = bp[(ka + 1) * BSZ + mn];
        acc0 = __builtin_amdgcn_wmma_f32_16x16x4_f32(false, a, false, b0,
                                                     (short)0, acc0, false, false);
        acc1 = __builtin_amdgcn_wmma_f32_16x16x4_f32(false, a, false, b1,
                                                     (short)0, acc1, false, false);
    }
    int rbase = m0 + ((lane >> 4) << 3);
    float sc = s_c[mn];
#pragma unroll
    for (int v = 0; v < 8; ++v) {
        int p = rbase + v;
        if (p < PP) {
            int idx = p * BSZ + mn;
            basesP[idx] = basesP[idx] * acc0[v] / (acc1[v] * sc + EPSV);
        }
    }
}

// ---------------- OUT: xrec[b][d][n] = bp[d]*coef[n] ----------------

__global__ __launch_bounds__(256) void k_out(const float* __restrict__ bp,
                                             const float* __restrict__ coef,
                                             float* __restrict__ out) {
    __shared__ float lbp[DD];
    int b = blockIdx.y, tid = threadIdx.x;
    if (tid < DD) lbp[tid] = bp[tid * BSZ + b];
    __syncthreads();
    int n = blockIdx.x * 256 + tid;
    float c = coef[(size_t)b * NN + n];
    float* ob = out + (size_t)b * DD * NN;
#pragma unroll 4
    for (int d = 0; d < DD; ++d)
        ob[(size_t)d * NN + n] = lbp[d] * c;
}

// ---------------- host ----------------

extern "C" void kernel_launch(void* const* d_in, const int* in_sizes, int n_in,
                              void* d_out, int out_size, void* d_ws, size_t ws_size,
                              hipStream_t stream) {
    const float* x      = (const float*)d_in[0];  // (16,16,160,36,36) == (16,160,20736) flat
    const float* bases0 = (const float*)d_in[1];  // (16,447,1)
    float* out = (float*)d_out;

    float* ws     = (float*)d_ws;
    float* rbf    = ws;                       // 160*448
    float* basesP = rbf    + DD * PPQ;        // 448*16   ([p][b])
    float* coef   = basesP + PPQ * BSZ;       // 16*20736 ([b][n])
    float* bp     = coef   + (size_t)BSZ * NN;// 160*16   ([d][b])
    float* xc     = bp     + DD * BSZ;        // 160*16   ([d][b])
    float* s_c    = xc     + DD * BSZ;        // 16

    k_rbf  <<<(DD * PPQ + 255) / 256, 256, 0, stream>>>(rbf);
    k_initb<<<(PPQ * BSZ + 255) / 256, 256, 0, stream>>>(bases0, basesP);
    k_initc<<<((size_t)BSZ * NN + 255) / 256, 256, 0, stream>>>(coef);

    for (int step = 0; step < 4; ++step) {
        k_s1<<<1, 320, 0, stream>>>(rbf, basesP, bp);
        k_nc<<<dim3(NN / 256, BSZ), 256, 0, stream>>>(x, bp, coef);
        k_sc<<<BSZ, 256, 0, stream>>>(coef, s_c);
        k_xc<<<dim3(DD, BSZ), 256, 0, stream>>>(x, coef, xc);
        k_s2<<<1, 896, 0, stream>>>(rbf, xc, bp, s_c, basesP);
    }
    // final differentiable coef pass + reconstruction
    k_s1 <<<1, 320, 0, stream>>>(rbf, basesP, bp);
    k_nc <<<dim3(NN / 256, BSZ), 256, 0, stream>>>(x, bp, coef);
    k_out<<<dim3(NN / 256, BSZ), 256, 0, stream>>>(bp, coef, out);
}